// Linears_9294309228757
// MI455X (gfx1250) — compile-verified
//
#include <hip/hip_runtime.h>

typedef float v2f __attribute__((ext_vector_type(2)));
typedef float v8f __attribute__((ext_vector_type(8)));

#define DIM        128
#define TWOD       256
#define KSTEPS     64          // 256 / 4
#define NT         8           // 128 / 16
#define WAVES_PB   8
#define EDGES_PW   32          // two 16-edge M-tiles per wave
#define EDGES_PB   (WAVES_PB * EDGES_PW)   // 256 edges per block

__global__ __launch_bounds__(256)
void edge_mlp_wmma_f32(const float* __restrict__ z_drug,
                       const float* __restrict__ z_dis,
                       const int*   __restrict__ row,
                       const int*   __restrict__ col,
                       const float* __restrict__ W1,
                       const float* __restrict__ b1,
                       const float* __restrict__ W2,
                       const float* __restrict__ b2,
                       float*       __restrict__ out,
                       int E)
{
    // W1 (256x128 f32, 128 KB) staged in WMMA B-fragment order:
    // frag(kk,nt): lane L<16  -> { W1[4kk+0][16nt+L],    W1[4kk+1][16nt+L] }
    //              lane L>=16 -> { W1[4kk+2][16nt+L-16], W1[4kk+3][16nt+L-16] }
    __shared__ float smW1[KSTEPS * NT * 32 * 2];   // 32768 floats

    const int tid  = threadIdx.x;
    const int lane = tid & 31;
    const int wave = tid >> 5;

    // ---- stage + swizzle W1 (coalesced global reads) ----
    for (int i = tid; i < TWOD * DIM; i += (int)blockDim.x) {
        const int k    = i >> 7;          // 0..255
        const int n    = i & (DIM - 1);   // 0..127
        const int kk   = k >> 2;
        const int krem = k & 3;
        const int hi   = krem >> 1;
        const int c    = krem & 1;
        const int l    = (n & 15) + (hi << 4);
        const int nt   = n >> 4;
        smW1[(((kk * NT + nt) * 32) + l) * 2 + c] = W1[i];
    }
    __syncthreads();
    const v2f* smB = (const v2f*)smW1;

    // ---- per-wave edge tile ----
    const int e0   = blockIdx.x * EDGES_PB + wave * EDGES_PW;
    const int mrow = lane & 15;            // M row within tile / N lane in epilogue
    const int khi  = (lane >> 4) << 1;     // 0 or 2 : K offset for hi half-wave

    int eA = e0 + mrow;                    // M-tile 0 edge
    int eB = eA + 16;                      // M-tile 1 edge
    eA = eA < E ? eA : E - 1;
    eB = eB < E ? eB : E - 1;

    const float* pA = z_drug + (size_t)row[eA] * DIM;
    const float* qA = z_dis  + (size_t)col[eA] * DIM;
    const float* pB = z_drug + (size_t)row[eB] * DIM;
    const float* qB = z_dis  + (size_t)col[eB] * DIM;

    const v8f vzero = {0.f, 0.f, 0.f, 0.f, 0.f, 0.f, 0.f, 0.f};
    v8f acc0[NT], acc1[NT];
    #pragma unroll
    for (int nt = 0; nt < NT; ++nt) { acc0[nt] = vzero; acc1[nt] = vzero; }

    // ---- K = 0..127 : drug half of the concat ----
    for (int kk = 0; kk < 32; ++kk) {
        v2f a0 = *(const v2f*)(pA + kk * 4 + khi);
        v2f a1 = *(const v2f*)(pB + kk * 4 + khi);
        #pragma unroll
        for (int nt = 0; nt < NT; ++nt) {
            v2f b = smB[(kk * NT + nt) * 32 + lane];
            acc0[nt] = __builtin_amdgcn_wmma_f32_16x16x4_f32(
                false, a0, false, b, (short)0, acc0[nt], false, false);
            acc1[nt] = __builtin_amdgcn_wmma_f32_16x16x4_f32(
                false, a1, false, b, (short)0, acc1[nt], false, false);
        }
    }
    // ---- K = 128..255 : disease half of the concat ----
    for (int kk = 0; kk < 32; ++kk) {
        v2f a0 = *(const v2f*)(qA + kk * 4 + khi);
        v2f a1 = *(const v2f*)(qB + kk * 4 + khi);
        #pragma unroll
        for (int nt = 0; nt < NT; ++nt) {
            v2f b = smB[((kk + 32) * NT + nt) * 32 + lane];
            acc0[nt] = __builtin_amdgcn_wmma_f32_16x16x4_f32(
                false, a0, false, b, (short)0, acc0[nt], false, false);
            acc1[nt] = __builtin_amdgcn_wmma_f32_16x16x4_f32(
                false, a1, false, b, (short)0, acc1[nt], false, false);
        }
    }

    // ---- epilogue: out[e] = relu(h + b1) . W2 + b2 ----
    // C/D layout: acc[nt][j] at lane L = h[M = j + 8*(L>=16)][n = 16*nt + (L&15)]
    float bb[NT], ww[NT];
    #pragma unroll
    for (int nt = 0; nt < NT; ++nt) {
        bb[nt] = b1[nt * 16 + mrow];
        ww[nt] = W2[nt * 16 + mrow];
    }
    const float beta = b2[0];
    const int   moff = (lane >> 4) << 3;   // hi half holds M = j + 8

    #pragma unroll
    for (int m = 0; m < 2; ++m) {
        #pragma unroll
        for (int j = 0; j < 8; ++j) {
            float s = 0.0f;
            #pragma unroll
            for (int nt = 0; nt < NT; ++nt) {
                float h = (m == 0 ? acc0[nt][j] : acc1[nt][j]) + bb[nt];
                h = h > 0.0f ? h : 0.0f;
                s = fmaf(h, ww[nt], s);
            }
            // reduce the 16 N-lanes of each half-wave
            s += __shfl_xor(s, 1, 32);
            s += __shfl_xor(s, 2, 32);
            s += __shfl_xor(s, 4, 32);
            s += __shfl_xor(s, 8, 32);
            if (mrow == 0) {
                const int e = e0 + m * 16 + j + moff;
                if (e < E) out[e] = s + beta;
            }
        }
    }
}

extern "C" void kernel_launch(void* const* d_in, const int* in_sizes, int n_in,
                              void* d_out, int out_size, void* d_ws, size_t ws_size,
                              hipStream_t stream) {
    const float* z_drug = (const float*)d_in[0];
    const float* z_dis  = (const float*)d_in[1];
    const int*   row    = (const int*)  d_in[2];
    const int*   col    = (const int*)  d_in[3];
    const float* W1     = (const float*)d_in[4];
    const float* b1     = (const float*)d_in[5];
    const float* W2     = (const float*)d_in[6];
    const float* b2     = (const float*)d_in[7];
    float* out = (float*)d_out;

    const int E = in_sizes[2];                      // number of edges
    const int blocks = (E + EDGES_PB - 1) / EDGES_PB;
    edge_mlp_wmma_f32<<<blocks, 256, 0, stream>>>(
        z_drug, z_dis, row, col, W1, b1, W2, b2, out, E);
}